// GlobalSSM_55095840473244
// MI455X (gfx1250) — compile-verified
//
#include <hip/hip_runtime.h>
#include <math.h>

typedef __attribute__((ext_vector_type(2))) float v2f;
typedef __attribute__((ext_vector_type(8))) float v8f;

#define D_MODEL   1024
#define STATE_DIM 64
#define BATCH     2
#define SEQ       1024
#define ROWS      (BATCH * SEQ)   // 2048

// ---------------------------------------------------------------------------
// helpers
// ---------------------------------------------------------------------------
__device__ __forceinline__ float softplusf(float v) {
    // jax.nn.softplus = log1p(exp(x)); for large x it is x
    return (v > 20.0f) ? v : log1pf(__expf(v));
}

__device__ __forceinline__ float wave_reduce_sum(float v) {
#pragma unroll
    for (int off = 16; off > 0; off >>= 1)
        v += __shfl_xor(v, off, 32);
    return v;
}

__device__ __forceinline__ float wave_reduce_max(float v) {
#pragma unroll
    for (int off = 16; off > 0; off >>= 1)
        v = fmaxf(v, __shfl_xor(v, off, 32));
    return v;
}

// byte offset of a shared-memory object inside the wave's LDS allocation
__device__ __forceinline__ uint32_t lds_off(const void* p) {
    return (uint32_t)(uintptr_t)(__attribute__((address_space(3))) const void*)p;
}

// Async DMA: 16 bytes global -> LDS, tracked by ASYNCcnt (per-lane addresses).
// Inline asm: portable across ROCm7.2/amdgpu-toolchain builtin-arity split.
__device__ __forceinline__ void async_cp_b128(uint32_t lds_byte, const void* gp) {
    asm volatile("global_load_async_to_lds_b128 %0, %1, off"
                 :: "v"(lds_byte), "v"((unsigned long long)(uintptr_t)gp)
                 : "memory");
}

__device__ __forceinline__ void wait_async0() {
    asm volatile("s_wait_asynccnt 0x0" ::: "memory");
}

// ---------------------------------------------------------------------------
// WMMA fp32 GEMM with async-LDS double-buffered staging.
//   out[r][c] = act( sum_k X[r][k] * W[c][k] )   (out = X @ W^T)
// Workgroup (256 thr = 8 waves) computes a 32(M) x 64(N) macro-tile; each wave
// owns one 16x16 sub-tile (2x4 arrangement) driven by V_WMMA_F32_16X16X4_F32.
// Per K-chunk (KC=32): X 32x32 and W 64x32 tiles are DMA'd into LDS with
// global_load_async_to_lds_b128 (3 ops/thread), double-buffered so the DMA of
// chunk c+1 overlaps the 8 WMMAs of chunk c. Rows padded to 36 floats:
// 16B-aligned rows and bank-conflict-free column reads (36*r mod 64 distinct
// over any 16 consecutive rows).
// ---------------------------------------------------------------------------
template <bool SOFTPLUS>
__global__ __launch_bounds__(256)
void gemm_xwt_async_wmma(const float* __restrict__ X,
                         const float* __restrict__ W,
                         float* __restrict__ out,
                         int R, int C, int K)
{
    (void)R;
    const int tid  = threadIdx.x;
    const int lane = tid & 31;
    const int wv   = tid >> 5;

    const int tilesN = C >> 6;                 // C / 64
    const int tM     = blockIdx.x / tilesN;
    const int tN     = blockIdx.x - tM * tilesN;
    const int row0g  = tM << 5;                // 32-row band
    const int col0g  = tN << 6;                // 64-col band

    __shared__ float lsX[2][32 * 36];          // 2 x 4.5 KB
    __shared__ float lsW[2][64 * 36];          // 2 x 9 KB

    // ---- DMA mapping: 256 threads move X(32x32)=256 b128 + W(64x32)=512 b128
    const int  ldRow = tid >> 3;               // 0..31
    const int  ldSeg = tid & 7;                // 0..7 (x4 floats)
    const float* xg  = X + (size_t)(row0g + ldRow) * K + (ldSeg << 2);
    const float* wg0 = W + (size_t)(col0g + ldRow) * K + (ldSeg << 2);
    const float* wg1 = wg0 + (size_t)32 * K;
    const uint32_t xB  = (uint32_t)((ldRow * 36 + (ldSeg << 2)) * 4);
    const uint32_t wB0 = xB;
    const uint32_t wB1 = (uint32_t)(((ldRow + 32) * 36 + (ldSeg << 2)) * 4);

    auto issue = [&](int c, int bf) {
        const int kc = c << 5;
        async_cp_b128(lds_off(&lsX[bf][0]) + xB,  xg  + kc);
        async_cp_b128(lds_off(&lsW[bf][0]) + wB0, wg0 + kc);
        async_cp_b128(lds_off(&lsW[bf][0]) + wB1, wg1 + kc);
    };

    // ---- wave -> 16x16 sub-tile
    const int rowL = (wv >> 2) << 4;           // 0 / 16
    const int colL = (wv & 3) << 4;            // 0 / 16 / 32 / 48
    const int half = lane >> 4;                // K-half select
    const int l15  = lane & 15;

    v8f acc0 = {};
    v8f acc1 = {};

    issue(0, 0);
    wait_async0();
    __syncthreads();

    const int nCh = K >> 5;
    for (int c = 0; c < nCh; ++c) {
        const int bf = c & 1;
        if (c + 1 < nCh) issue(c + 1, bf ^ 1); // overlap DMA with WMMA

        const float* aB = &lsX[bf][(rowL + l15) * 36 + (half << 1)];
        const float* bX = &lsW[bf][(colL + l15) * 36 + (half << 1)];
#pragma unroll
        for (int kk = 0; kk < 32; kk += 8) {
            v2f a0 = *(const v2f*)(aB + kk);
            v2f b0 = *(const v2f*)(bX + kk);
            v2f a1 = *(const v2f*)(aB + kk + 4);
            v2f b1 = *(const v2f*)(bX + kk + 4);
            acc0 = __builtin_amdgcn_wmma_f32_16x16x4_f32(
                       false, a0, false, b0, (short)0, acc0, false, false);
            acc1 = __builtin_amdgcn_wmma_f32_16x16x4_f32(
                       false, a1, false, b1, (short)0, acc1, false, false);
        }
        wait_async0();      // own chunk-(c+1) DMA done
        __syncthreads();    // everyone done reading buf bf + everyone's DMA done
    }
    v8f acc = acc0 + acc1;

    // D layout: VGPR i -> row (rowL + 8*half + i), col (colL + l15)
    float* o = out + (size_t)(row0g + rowL + (half << 3)) * C + col0g + colL + l15;
#pragma unroll
    for (int i = 0; i < 8; ++i) {
        float v = acc[i];
        if (SOFTPLUS) v = softplusf(v);
        o[(size_t)i * C] = v;
    }
}

// ---------------------------------------------------------------------------
// Selective scan. One wave per (b,d) channel; 2 states per lane (64/32).
// Only the final-step y is needed downstream, so no per-step cross-lane
// reduction -- h lives entirely in VGPRs for all 1024 steps.
// ---------------------------------------------------------------------------
__global__ void scan_kernel(const float* __restrict__ x,
                            const float* __restrict__ delta,
                            const float* __restrict__ Bsel,
                            const float* __restrict__ A_log,
                            const float* __restrict__ Clast,
                            float* __restrict__ y_last)
{
    const int lane = threadIdx.x & 31;
    const int wave = (blockIdx.x * blockDim.x + threadIdx.x) >> 5; // 0..2047
    const int b    = wave >> 10;
    const int d    = wave & (D_MODEL - 1);
    const int n0   = lane << 1;

    v2f  Av = *(const v2f*)(A_log + (size_t)d * STATE_DIM + n0);
    const float A0 = -__expf(Av.x);
    const float A1 = -__expf(Av.y);

    const float* xb = x     + (size_t)b * SEQ * D_MODEL + d;
    const float* db = delta + (size_t)b * SEQ * D_MODEL + d;
    const float* Bb = Bsel  + (size_t)b * SEQ * STATE_DIM + n0;

    float h0 = 0.0f, h1 = 0.0f;
    for (int t = 0; t < SEQ; ++t) {
        __builtin_prefetch(db + (size_t)(t + 8) * D_MODEL, 0, 3);  // near-scope
        __builtin_prefetch(xb + (size_t)(t + 8) * D_MODEL, 0, 3);
        const float dt = db[(size_t)t * D_MODEL];
        const float xt = xb[(size_t)t * D_MODEL];
        const v2f   bt = *(const v2f*)(Bb + (size_t)t * STATE_DIM);
        const float u  = dt * xt;
        h0 = h0 * __expf(dt * A0) + u * bt.x;
        h1 = h1 * __expf(dt * A1) + u * bt.y;
    }

    const v2f cv = *(const v2f*)(Clast + b * STATE_DIM + n0);
    float y = h0 * cv.x + h1 * cv.y;
    y = wave_reduce_sum(y);
    if (lane == 0) y_last[b * D_MODEL + d] = y;
}

// ---------------------------------------------------------------------------
// Generic wave-per-dot kernel: dot i  (outer = i / inner, in = i % inner)
//   a = A + outer*aOut + in*aIn ; b = Bv + outer*bOut + in*bIn
// ---------------------------------------------------------------------------
__global__ void wave_dot_kernel(const float* __restrict__ A, size_t aOut, size_t aIn,
                                const float* __restrict__ Bv, size_t bOut, size_t bIn,
                                float* __restrict__ out, int inner, int nDots, int K)
{
    const int lane = threadIdx.x & 31;
    const int w    = (blockIdx.x * blockDim.x + threadIdx.x) >> 5;
    if (w >= nDots) return;
    const int outer = w / inner;
    const int in    = w - outer * inner;
    const float* a = A  + (size_t)outer * aOut + (size_t)in * aIn;
    const float* b = Bv + (size_t)outer * bOut + (size_t)in * bIn;
    float s = 0.0f;
    for (int k = lane; k < K; k += 32) s = fmaf(a[k], b[k], s);
    s = wave_reduce_sum(s);
    if (lane == 0) out[w] = s;
}

// ---------------------------------------------------------------------------
// LayerNorm over one 1024-row per block (only the last timestep matters).
// ---------------------------------------------------------------------------
__global__ void layernorm_kernel(const float* __restrict__ yin,
                                 const float* __restrict__ gamma,
                                 const float* __restrict__ beta,
                                 float* __restrict__ outp)
{
    __shared__ float redS[8], redS2[8];
    __shared__ float s_mu, s_rstd;
    const int b = blockIdx.x, tid = threadIdx.x;
    const float* row = yin + (size_t)b * D_MODEL;

    float v[4], s = 0.0f, s2 = 0.0f;
#pragma unroll
    for (int j = 0; j < 4; ++j) {
        float t = row[tid + j * 256];
        v[j] = t; s += t; s2 += t * t;
    }
    s  = wave_reduce_sum(s);
    s2 = wave_reduce_sum(s2);
    const int lane = tid & 31, wv = tid >> 5;
    if (lane == 0) { redS[wv] = s; redS2[wv] = s2; }
    __syncthreads();
    if (tid == 0) {
        float ts = 0.0f, ts2 = 0.0f;
#pragma unroll
        for (int i = 0; i < 8; ++i) { ts += redS[i]; ts2 += redS2[i]; }
        const float mu  = ts / (float)D_MODEL;
        const float var = ts2 / (float)D_MODEL - mu * mu;
        s_mu = mu; s_rstd = rsqrtf(var + 1e-5f);
    }
    __syncthreads();
#pragma unroll
    for (int j = 0; j < 4; ++j) {
        const int i = tid + j * 256;
        outp[(size_t)b * D_MODEL + i] = (v[j] - s_mu) * s_rstd * gamma[i] + beta[i];
    }
}

// ---------------------------------------------------------------------------
// Softmax over SEQ with temperature 0.5 (logit scale 2.0); one block per batch.
// ---------------------------------------------------------------------------
__global__ void softmax_kernel(const float* __restrict__ raw,
                               float* __restrict__ imp)
{
    __shared__ float red[8];
    __shared__ float s_max, s_sum;
    const int b = blockIdx.x, tid = threadIdx.x;
    const float* row = raw + (size_t)b * SEQ;

    float l[4], m = -3.0e38f;
#pragma unroll
    for (int j = 0; j < 4; ++j) {
        l[j] = row[tid + j * 256] * 2.0f;   // raw / TEMPERATURE
        m = fmaxf(m, l[j]);
    }
    m = wave_reduce_max(m);
    const int lane = tid & 31, wv = tid >> 5;
    if (lane == 0) red[wv] = m;
    __syncthreads();
    if (tid == 0) {
        float tm = red[0];
#pragma unroll
        for (int i = 1; i < 8; ++i) tm = fmaxf(tm, red[i]);
        s_max = tm;
    }
    __syncthreads();

    float e[4], s = 0.0f;
#pragma unroll
    for (int j = 0; j < 4; ++j) { e[j] = __expf(l[j] - s_max); s += e[j]; }
    s = wave_reduce_sum(s);
    if (lane == 0) red[wv] = s;
    __syncthreads();
    if (tid == 0) {
        float ts = 0.0f;
#pragma unroll
        for (int i = 0; i < 8; ++i) ts += red[i];
        s_sum = ts;
    }
    __syncthreads();
    const float inv = 1.0f / s_sum;
#pragma unroll
    for (int j = 0; j < 4; ++j)
        imp[(size_t)b * SEQ + tid + j * 256] = e[j] * inv;
}

// ---------------------------------------------------------------------------
// launcher
// ---------------------------------------------------------------------------
extern "C" void kernel_launch(void* const* d_in, const int* in_sizes, int n_in,
                              void* d_out, int out_size, void* d_ws, size_t ws_size,
                              hipStream_t stream)
{
    (void)in_sizes; (void)n_in; (void)out_size; (void)ws_size;
    const float* x       = (const float*)d_in[0];
    const float* A_log   = (const float*)d_in[1];
    const float* W_delta = (const float*)d_in[2];
    const float* W_B     = (const float*)d_in[3];
    const float* W_C     = (const float*)d_in[4];
    const float* W_proj  = (const float*)d_in[5];
    const float* gamma   = (const float*)d_in[6];
    const float* beta    = (const float*)d_in[7];

    float* out = (float*)d_out;             // [0..2047]=importance, [2048..4095]=raw
    float* ws  = (float*)d_ws;

    float* delta  = ws;                                        // 2048*1024
    float* Bsel   = delta  + (size_t)ROWS * D_MODEL;           // 2048*64
    float* Clast  = Bsel   + (size_t)ROWS * STATE_DIM;         // 2*64
    float* ylast  = Clast  + BATCH * STATE_DIM;                // 2*1024
    float* normed = ylast  + BATCH * D_MODEL;                  // 2*1024
    float* hproj  = normed + BATCH * D_MODEL;                  // 2*1024
    float* raw    = out + BATCH * SEQ;                         // second output

    // 1) delta = softplus(x @ W_delta^T): (2048/32)x(1024/64) = 1024 workgroups
    gemm_xwt_async_wmma<true><<<(ROWS / 32) * (D_MODEL / 64), 256, 0, stream>>>(
        x, W_delta, delta, ROWS, D_MODEL, D_MODEL);
    // 2) B_sel = x @ W_B^T: (2048/32)x(64/64) = 64 workgroups
    gemm_xwt_async_wmma<false><<<(ROWS / 32) * (STATE_DIM / 64), 256, 0, stream>>>(
        x, W_B, Bsel, ROWS, STATE_DIM, D_MODEL);
    // 3) C needed only at t = SEQ-1:  Clast[b][n] = dot(x[b][S-1][:], W_C[n][:])
    wave_dot_kernel<<<(BATCH * STATE_DIM * 32) / 256, 256, 0, stream>>>(
        x + (size_t)(SEQ - 1) * D_MODEL, (size_t)SEQ * D_MODEL, 0,
        W_C, 0, D_MODEL,
        Clast, STATE_DIM, BATCH * STATE_DIM, D_MODEL);
    // 4) sequential scan -> y at t = SEQ-1
    scan_kernel<<<(BATCH * D_MODEL) / 8, 256, 0, stream>>>(
        x, delta, Bsel, A_log, Clast, ylast);
    // 5) LayerNorm of the final-step row
    layernorm_kernel<<<BATCH, 256, 0, stream>>>(ylast, gamma, beta, normed);
    // 6) h_proj[b][j] = dot(normed[b][:], W_proj[j][:])
    wave_dot_kernel<<<(BATCH * D_MODEL * 32) / 256, 256, 0, stream>>>(
        normed, D_MODEL, 0,
        W_proj, 0, D_MODEL,
        hproj, D_MODEL, BATCH * D_MODEL, D_MODEL);
    // 7) raw[b][s] = dot(x[b][s][:], h_proj[b][:])
    wave_dot_kernel<<<(BATCH * SEQ * 32) / 256, 256, 0, stream>>>(
        x, (size_t)SEQ * D_MODEL, D_MODEL,
        hproj, D_MODEL, 0,
        raw, SEQ, BATCH * SEQ, D_MODEL);
    // 8) importance = softmax(raw / 0.5)
    softmax_kernel<<<BATCH, 256, 0, stream>>>(raw, out);
}